// LocalContrastiveLoss_61890478735388
// MI455X (gfx1250) — compile-verified
//
#include <hip/hip_runtime.h>

// Problem constants (from reference): B=8, E=64, H=W=256, K=8, NPOS=4
#define B_    8
#define E_    64
#define HW_   65536
#define K_    8
#define S_    4
#define TILE_ 2048                 // pixels per pass-1 block
#define CPI_  (HW_ / TILE_)        // 32 chunks per image
#define NBLK_ (B_ * CPI_)          // 256 pass-1 blocks
#define REP_  4                    // replicated LDS accumulators (conflict spread)
#define EPAD_ 68                   // bank-padded E stride
#define EPSF  1e-8f

typedef float v2f __attribute__((ext_vector_type(2)));
typedef float v4f __attribute__((ext_vector_type(4)));
typedef float v8f __attribute__((ext_vector_type(8)));

__device__ __forceinline__ v4f ntload4(const float* p) {
  return __builtin_nontemporal_load((const v4f*)p);  // streaming: TH_NT
}

// ---------------------------------------------------------------------------
// Pass 1: streaming masked reduction.  Each block owns TILE_ pixels of one
// image: decode one-hot masks -> per-pixel labels (LDS), then stream the 64
// embedding channels (float4, non-temporal) accumulating into replicated
// LDS [K][E] partial sums via ds_add_f32.  Deterministic per-block partials.
// ---------------------------------------------------------------------------
__global__ void __launch_bounds__(256) lcl_pass1(
    const float* __restrict__ emb, const float* __restrict__ mask,
    float* __restrict__ psum, float* __restrict__ pcnt)
{
  __shared__ unsigned lblw[TILE_ / 4];       // 4 packed u8 labels per word
  __shared__ float acc[REP_][K_][EPAD_];
  __shared__ float ccnt[K_];

  const int blk = blockIdx.x;
  const int b   = blk / CPI_;
  const int n0  = (blk % CPI_) * TILE_;
  const int t   = threadIdx.x;

  for (int i = t; i < REP_ * K_ * EPAD_; i += 256) ((float*)acc)[i] = 0.f;
  if (t < K_) ccnt[t] = 0.f;
  __syncthreads();

  // Phase A: labels + counts (masks are exact 0.0/1.0 one-hot)
  int cloc[K_];
  #pragma unroll
  for (int k = 0; k < K_; ++k) cloc[k] = 0;

  #pragma unroll
  for (int g = 0; g < TILE_ / 1024; ++g) {
    const int pw = g * 256 + t;        // word index within tile
    const int p  = n0 + pw * 4;        // pixel index
    unsigned l0 = 0, l1 = 0, l2 = 0, l3 = 0;
    #pragma unroll
    for (int k = 0; k < K_; ++k) {
      v4f m = ntload4(mask + (size_t)(b * K_ + k) * HW_ + p);
      if (m.x > 0.5f) l0 = (unsigned)k;
      if (m.y > 0.5f) l1 = (unsigned)k;
      if (m.z > 0.5f) l2 = (unsigned)k;
      if (m.w > 0.5f) l3 = (unsigned)k;
      cloc[k] += (m.x > 0.5f) + (m.y > 0.5f) + (m.z > 0.5f) + (m.w > 0.5f);
    }
    lblw[pw] = l0 | (l1 << 8) | (l2 << 16) | (l3 << 24);
  }
  #pragma unroll
  for (int k = 0; k < K_; ++k) atomicAdd(&ccnt[k], (float)cloc[k]);
  __syncthreads();

  // Phase B: stream embeddings, scatter-accumulate by label (ds_add_f32)
  const int wave = t >> 5, lane = t & 31;
  const int cp   = lane >> 3;          // 4 accumulator copies spread conflicts
  #pragma unroll
  for (int ei = 0; ei < 8; ++ei) {
    const int e = wave * 8 + ei;       // wave owns 8 channels
    const float* base = emb + (size_t)(b * E_ + e) * HW_ + n0;
    for (int j = lane; j < TILE_ / 4; j += 32) {
      v4f v = ntload4(base + j * 4);   // coalesced 512B per wave-instr
      unsigned pk = lblw[j];
      atomicAdd(&acc[cp][pk & 7u][e], v.x);
      atomicAdd(&acc[cp][(pk >> 8) & 7u][e], v.y);
      atomicAdd(&acc[cp][(pk >> 16) & 7u][e], v.z);
      atomicAdd(&acc[cp][(pk >> 24) & 7u][e], v.w);
    }
  }
  __syncthreads();

  float* outp = psum + (size_t)blk * (K_ * E_);
  for (int i = t; i < K_ * E_; i += 256) {
    const int k = i >> 6, e = i & 63;
    outp[i] = acc[0][k][e] + acc[1][k][e] + acc[2][k][e] + acc[3][k][e];
  }
  if (t < K_) pcnt[blk * K_ + t] = ccnt[t];
}

// ---------------------------------------------------------------------------
// Pass 2 helper: extract per-row logsumexp - s_pos from a 16x16 f32 C-fragment.
// C layout: lanes 0-15 hold column N=lane for rows M=0..7 in VGPRs 0..7,
// lanes 16-31 hold rows M=8..15.  Classes 8..15 are zero-pad columns and sit
// in their own xor-8 shuffle groups (lanes 8-15 / 24-31) -> never mixed in.
// ---------------------------------------------------------------------------
__device__ __forceinline__ float tile_loss(v8f c, int l, int sbase) {
  float lacc = 0.f;
  const int half = l >> 4, col = l & 15;
  #pragma unroll
  for (int r = 0; r < 8; ++r) {
    float v  = c[r];
    float mx = v;
    mx = fmaxf(mx, __shfl_xor(mx, 1));
    mx = fmaxf(mx, __shfl_xor(mx, 2));
    mx = fmaxf(mx, __shfl_xor(mx, 4));
    float sm = expf(v - mx);
    sm += __shfl_xor(sm, 1);
    sm += __shfl_xor(sm, 2);
    sm += __shfl_xor(sm, 4);
    const float lse = mx + logf(sm);
    const int kk = (sbase + 8 * half + r) >> 2;   // positive class of this row
    lacc += (col == kk) ? (lse - v) : 0.f;
  }
  return lacc;
}

// ---------------------------------------------------------------------------
// Pass 2: one wave per image.  Reduce partial sums -> normalized class means,
// gather + normalize 32 sampled pixel embeddings (1/TEMP folded in), then
// sims = Zn (32x64) x MnT (64x16) via V_WMMA_F32_16X16X4_F32 (two 16-row
// tiles, 16 k-steps each), and the contrastive loss terms from the fragment.
// ---------------------------------------------------------------------------
__global__ void __launch_bounds__(32) lcl_pass2(
    const float* __restrict__ emb, const int* __restrict__ pos,
    const float* __restrict__ psum, const float* __restrict__ pcnt,
    float* __restrict__ wloss)
{
  __shared__ float mn[16][E_];       // rows 8..15 zero-pad for WMMA N=16
  __shared__ float zn[32][E_];
  __shared__ float cnt[K_];

  const int b = blockIdx.x;
  const int l = threadIdx.x;

  if (l < K_) {
    float c = 0.f;
    for (int p = 0; p < CPI_; ++p) c += pcnt[(b * CPI_ + p) * K_ + l];
    cnt[l] = fmaxf(c, 1.0f);
  }
  __syncthreads();

  // means (sum partials / count); pad classes 8..15 with zeros
  for (int idx = l; idx < 16 * E_; idx += 32) {
    const int k = idx >> 6;
    float val = 0.f;
    if (k < K_) {
      float s = 0.f;
      for (int p = 0; p < CPI_; ++p)
        s += psum[(size_t)(b * CPI_ + p) * (K_ * E_) + idx];
      val = s / cnt[k];
    }
    mn[k][idx & 63] = val;
  }
  __syncthreads();

  // normalize means: mn = mean / max(||mean||, EPS)
  #pragma unroll
  for (int k = 0; k < K_; ++k) {
    const float a0 = mn[k][l], a1 = mn[k][l + 32];
    float p = a0 * a0 + a1 * a1;
    #pragma unroll
    for (int m = 1; m < 32; m <<= 1) p += __shfl_xor(p, m);
    const float inv = 1.0f / fmaxf(sqrtf(p), EPSF);
    mn[k][l] = a0 * inv;
    mn[k][l + 32] = a1 * inv;
  }
  __syncthreads();

  // gather + normalize sampled pixel embeddings; fold 1/TEMP = 5 into z
  const int pix = pos[b * (K_ * S_) + l];      // lane l = sample (k*4+s)
  float n2 = 0.f;
  for (int e = 0; e < E_; ++e) {
    const float v = emb[(size_t)(b * E_ + e) * HW_ + pix];
    zn[l][e] = v;
    n2 += v * v;
  }
  const float sc = 5.0f / fmaxf(sqrtf(n2), EPSF);
  for (int e = 0; e < E_; ++e) zn[l][e] *= sc;
  __syncthreads();

  // sims via WMMA f32 16x16x4.  A 16x4: lanes 0-15 -> K=e0,e0+1 for M=lane;
  // lanes 16-31 -> K=e0+2,e0+3.  B 4x16 mirrored; C: lane=column, vgpr=row.
  v8f acc0 = {}; v8f acc1 = {};
  const int half = l >> 4, r = l & 15;
  for (int e0 = 0; e0 < E_; e0 += 4) {
    const int eo = e0 + 2 * half;
    v2f a0 = *(const v2f*)&zn[r][eo];          // samples 0..15
    v2f a1 = *(const v2f*)&zn[16 + r][eo];     // samples 16..31
    v2f bf = *(const v2f*)&mn[r][eo];          // class column (zeros for 8..15)
    acc0 = __builtin_amdgcn_wmma_f32_16x16x4_f32(false, a0, false, bf,
                                                 (short)0, acc0, false, false);
    acc1 = __builtin_amdgcn_wmma_f32_16x16x4_f32(false, a1, false, bf,
                                                 (short)0, acc1, false, false);
  }

  float lacc = tile_loss(acc0, l, 0) + tile_loss(acc1, l, 16);
  #pragma unroll
  for (int m = 1; m < 32; m <<= 1) lacc += __shfl_xor(lacc, m);
  if (l == 0) wloss[b] = lacc;
}

// ---------------------------------------------------------------------------
// Pass 3: deterministic final reduction: mean over all 256 (b,k,s) terms.
// ---------------------------------------------------------------------------
__global__ void lcl_pass3(const float* __restrict__ wloss,
                          float* __restrict__ out)
{
  if (threadIdx.x == 0) {
    float s = 0.f;
    for (int b = 0; b < B_; ++b) s += wloss[b];
    out[0] = s * (1.0f / 256.0f);
  }
}

extern "C" void kernel_launch(void* const* d_in, const int* in_sizes, int n_in,
                              void* d_out, int out_size, void* d_ws, size_t ws_size,
                              hipStream_t stream)
{
  const float* emb  = (const float*)d_in[0];   // [8,64,256,256] f32
  const float* mask = (const float*)d_in[1];   // [8,8,256,256]  f32
  const int*   pos  = (const int*)d_in[2];     // [8,8,4]        i32
  float* out = (float*)d_out;                  // scalar f32

  float* psum  = (float*)d_ws;                         // NBLK_*512 floats
  float* pcnt  = psum + (size_t)NBLK_ * (K_ * E_);     // NBLK_*8 floats
  float* wloss = pcnt + (size_t)NBLK_ * K_;            // 8 floats

  lcl_pass1<<<NBLK_, 256, 0, stream>>>(emb, mask, psum, pcnt);
  lcl_pass2<<<B_, 32, 0, stream>>>(emb, pos, psum, pcnt, wloss);
  lcl_pass3<<<1, 32, 0, stream>>>(wloss, out);

  (void)in_sizes; (void)n_in; (void)out_size; (void)ws_size;
}